// HGNN_38938173505545
// MI455X (gfx1250) — compile-verified
//
#include <hip/hip_runtime.h>
#include <stdint.h>

#ifndef __has_builtin
#define __has_builtin(x) 0
#endif

#define FEATS 128

// Vector-of-4-int pointee type the gfx1250 async-LDS builtins expect.
typedef int v4i __attribute__((ext_vector_type(4)));
typedef __attribute__((address_space(1))) v4i GV4;   // global int4*
typedef __attribute__((address_space(3))) v4i LV4;   // LDS int4*

#if __has_builtin(__builtin_amdgcn_global_load_async_to_lds_b128) && \
    __has_builtin(__builtin_amdgcn_global_store_async_from_lds_b128)
#define ASYNC_COPY 1
#else
#define ASYNC_COPY 0
#endif

__device__ __forceinline__ void wait_asynccnt0() {
#if __has_builtin(__builtin_amdgcn_s_wait_asynccnt)
  __builtin_amdgcn_s_wait_asynccnt(0);
#else
  asm volatile("s_wait_asynccnt 0" ::: "memory");
#endif
}

// ---------------------------------------------------------------------------
// Kernel 1: out = x  (residual init; 51.2 MB copy, 3.2M float4 chunks).
// Uses the CDNA5 async global<->LDS data movers:
// global_load_async_to_lds_b128 -> s_wait_asynccnt 0 ->
// global_store_async_from_lds_b128 (S_ENDPGM's implicit wait-idle covers the
// store). Falls back to a plain b128 VGPR copy if builtins are absent.
// ---------------------------------------------------------------------------
__global__ void hgnn_residual_copy(const float4* __restrict__ in,
                                   float4* __restrict__ out, int n4) {
  int t = blockIdx.x * blockDim.x + threadIdx.x;
#if ASYNC_COPY
  __shared__ float4 stage[256];
  LV4* lds = (LV4*)(&stage[threadIdx.x]);
  if (t < n4) {
    __builtin_amdgcn_global_load_async_to_lds_b128(
        (GV4*)(uintptr_t)(in + t), lds, 0, 0);
  }
  wait_asynccnt0();
  if (t < n4) {
    __builtin_amdgcn_global_store_async_from_lds_b128(
        (GV4*)(uintptr_t)(out + t), lds, 0, 0);
  }
#else
  if (t < n4) out[t] = in[t];
#endif
}

// ---------------------------------------------------------------------------
// Kernel 2: edge-parallel scatter-add. One wave32 per edge:
//  - one coalesced 512B row gather (global_load_b128 per lane, L2-resident)
//  - 4 native fp32 L2 atomics per lane (global_atomic_add_f32, no return)
// dst rows are random over 51 MB (L2-resident), so conflict rate ~0.
// ---------------------------------------------------------------------------
__global__ void hgnn_scatter_add(const float* __restrict__ x,
                                 const int* __restrict__ src,
                                 const int* __restrict__ dst,
                                 float* __restrict__ out, int n_edges) {
  int gtid = blockIdx.x * blockDim.x + threadIdx.x;
  int e    = gtid >> 5;          // one edge per wave32
  int lane = gtid & 31;          // 4 consecutive floats per lane
  if (e >= n_edges) return;

  int s = src[e];
  int d = dst[e];

  const float4 v = *((const float4*)(x + (size_t)s * FEATS) + lane);
  float* o = out + (size_t)d * FEATS + lane * 4;

  unsafeAtomicAdd(o + 0, v.x);
  unsafeAtomicAdd(o + 1, v.y);
  unsafeAtomicAdd(o + 2, v.z);
  unsafeAtomicAdd(o + 3, v.w);
}

extern "C" void kernel_launch(void* const* d_in, const int* in_sizes, int n_in,
                              void* d_out, int out_size, void* d_ws, size_t ws_size,
                              hipStream_t stream) {
  const float* x   = (const float*)d_in[0];
  const int*   src = (const int*)d_in[1];
  const int*   dst = (const int*)d_in[2];
  float*       out = (float*)d_out;

  const int n_edges = in_sizes[1];        // 2,000,000
  const int n4      = out_size / 4;       // 12.8M floats -> 3.2M float4

  // Phase 1: out = x (fully rewrites d_out each call -> replay-safe).
  {
    int blocks = (n4 + 255) / 256;
    hgnn_residual_copy<<<blocks, 256, 0, stream>>>(
        (const float4*)x, (float4*)out, n4);
  }

  // Phase 2: atomic scatter-add of gathered rows (stream-ordered after copy).
  {
    long long total = (long long)n_edges * 32;
    int blocks = (int)((total + 255) / 256);
    hgnn_scatter_add<<<blocks, 256, 0, stream>>>(x, src, dst, out, n_edges);
  }
}